// NeighborAggregator_46273977647740
// MI455X (gfx1250) — compile-verified
//
#include <hip/hip_runtime.h>
#include <hip/hip_bf16.h>

// Problem constants (match the reference).
#define N_DIM 16384
#define K_DIM 32
#define C_DIM 16384
#define TOTAL_NK (N_DIM * K_DIM)   // 524288

// Typed pointers for the gfx1250 async load-to-LDS builtin:
// param0 = global (AS1) int4*, param1 = LDS (AS3) int4*.
typedef int v4i_t __attribute__((ext_vector_type(4)));
typedef __attribute__((address_space(1))) v4i_t* g_v4i_p;
typedef __attribute__((address_space(3))) v4i_t* l_v4i_p;

// ---------------------------------------------------------------------------
// Kernel 1: zero the accumulation buffer (reduced half of d_out).
// ---------------------------------------------------------------------------
__global__ __launch_bounds__(256) void na_zero_kernel(float* __restrict__ p) {
  p[blockIdx.x * 256 + threadIdx.x] = 0.0f;
}

// ---------------------------------------------------------------------------
// Kernel 2: gather + multiply + scatter-add.
// 4 entries per thread: int4 index load + float4 value load (coalesced),
// 4 independent row-gathers, 4 native f32 global atomics.
// ---------------------------------------------------------------------------
__global__ __launch_bounds__(256) void na_scatter_kernel(
    const float* __restrict__ input,     // (N, C)
    const int* __restrict__ indices,     // (N, K) int32
    const float* __restrict__ values,    // (N, K)
    float* __restrict__ reduced) {       // (C,)
  const int t = blockIdx.x * 256 + threadIdx.x;   // 0 .. TOTAL_NK/4 - 1
  const int g = t * 4;                            // first entry index

  const int4   iv = *reinterpret_cast<const int4*>(indices + g);
  const float4 vv = *reinterpret_cast<const float4*>(values + g);

  // g is a multiple of 4 and K=32, so all 4 entries share the same row n = g>>5.
  const float* row = input + (size_t)(g >> 5) * (size_t)C_DIM;

  const unsigned c0 = (unsigned)iv.x, c1 = (unsigned)iv.y;
  const unsigned c2 = (unsigned)iv.z, c3 = (unsigned)iv.w;

  const float d0 = row[c0];
  const float d1 = row[c1];
  const float d2 = row[c2];
  const float d3 = row[c3];

#if defined(__HIP_PLATFORM_AMD__) || defined(__HIP_PLATFORM_HCC__)
  unsafeAtomicAdd(&reduced[c0], vv.x * d0);   // -> global_atomic_add_f32 (no CAS loop)
  unsafeAtomicAdd(&reduced[c1], vv.y * d1);
  unsafeAtomicAdd(&reduced[c2], vv.z * d2);
  unsafeAtomicAdd(&reduced[c3], vv.w * d3);
#else
  atomicAdd(&reduced[c0], vv.x * d0);
  atomicAdd(&reduced[c1], vv.y * d1);
  atomicAdd(&reduced[c2], vv.z * d2);
  atomicAdd(&reduced[c3], vv.w * d3);
#endif
}

// ---------------------------------------------------------------------------
// Kernel 3: softmax over reduced[0..C) -> alpha[0..C).
// Single 1024-thread block (32 wave32 waves).
// Stage 64KB into LDS via gfx1250 async-load-to-LDS, pull 16 vals/thread into
// registers, then reuse the LDS buffer as cross-wave reduction scratch.
// ---------------------------------------------------------------------------
__global__ __launch_bounds__(1024) void na_softmax_kernel(
    const float* __restrict__ reduced, float* __restrict__ alpha) {
  __shared__ float s[C_DIM];            // exactly 64 KB
  const int tid  = threadIdx.x;
  const int lane = tid & 31;
  const int wave = tid >> 5;

  // ---- Stage reduced[] (64 KB) into LDS --------------------------------
#if __has_builtin(__builtin_amdgcn_global_load_async_to_lds_b128)
  #pragma unroll
  for (int i = 0; i < 4; ++i) {
    const int chunk = i * 1024 + tid;          // 16-byte chunk id, coalesced
    __builtin_amdgcn_global_load_async_to_lds_b128(
        (g_v4i_p)(reduced + chunk * 4),
        (l_v4i_p)(s + chunk * 4),
        /*offset=*/0, /*cpol=*/0);
  }
  #if __has_builtin(__builtin_amdgcn_s_wait_asynccnt)
  __builtin_amdgcn_s_wait_asynccnt(0);
  #else
  asm volatile("s_wait_asynccnt 0" ::: "memory");
  #endif
#else
  for (int i = tid; i < C_DIM / 4; i += 1024)
    reinterpret_cast<float4*>(s)[i] = reinterpret_cast<const float4*>(reduced)[i];
#endif
  __syncthreads();

  // ---- Pull this thread's 16 values into registers ----------------------
  float v[16];
  #pragma unroll
  for (int i = 0; i < 16; ++i) v[i] = s[tid + i * 1024];
  __syncthreads();   // all reads done before s is reused as scratch

  // ---- Block max (wave32 shuffle tree + 32-slot LDS scratch) ------------
  float m = v[0];
  #pragma unroll
  for (int i = 1; i < 16; ++i) m = fmaxf(m, v[i]);
  #pragma unroll
  for (int off = 16; off > 0; off >>= 1) m = fmaxf(m, __shfl_xor(m, off, 32));
  if (lane == 0) s[wave] = m;
  __syncthreads();
  if (wave == 0) {
    float x = s[lane];
    #pragma unroll
    for (int off = 16; off > 0; off >>= 1) x = fmaxf(x, __shfl_xor(x, off, 32));
    if (lane == 0) s[0] = x;
  }
  __syncthreads();
  const float bmax = s[0];
  __syncthreads();   // everyone has bmax before scratch is overwritten

  // ---- exp + block sum --------------------------------------------------
  float acc = 0.0f;
  #pragma unroll
  for (int i = 0; i < 16; ++i) {
    v[i] = __expf(v[i] - bmax);
    acc += v[i];
  }
  #pragma unroll
  for (int off = 16; off > 0; off >>= 1) acc += __shfl_xor(acc, off, 32);
  if (lane == 0) s[wave] = acc;
  __syncthreads();
  if (wave == 0) {
    float x = s[lane];
    #pragma unroll
    for (int off = 16; off > 0; off >>= 1) x += __shfl_xor(x, off, 32);
    if (lane == 0) s[0] = x;
  }
  __syncthreads();
  const float inv = 1.0f / s[0];

  // ---- Normalize & write alpha -----------------------------------------
  #pragma unroll
  for (int i = 0; i < 16; ++i) alpha[tid + i * 1024] = v[i] * inv;
}

// ---------------------------------------------------------------------------
// Launch: d_out = [ alpha (16384 f32) | reduced (16384 f32) ]
// ---------------------------------------------------------------------------
extern "C" void kernel_launch(void* const* d_in, const int* in_sizes, int n_in,
                              void* d_out, int out_size, void* d_ws, size_t ws_size,
                              hipStream_t stream) {
  (void)in_sizes; (void)n_in; (void)out_size; (void)d_ws; (void)ws_size;

  const float* input   = (const float*)d_in[0];
  const int*   indices = (const int*)d_in[1];   // integer inputs -> int32
  const float* values  = (const float*)d_in[2];
  // d_in[3] = k (scalar, always 32) -> compile-time constant

  float* alpha   = (float*)d_out;
  float* reduced = alpha + C_DIM;

  hipLaunchKernelGGL(na_zero_kernel, dim3(C_DIM / 256), dim3(256), 0, stream,
                     reduced);
  hipLaunchKernelGGL(na_scatter_kernel, dim3(TOTAL_NK / 4 / 256), dim3(256), 0,
                     stream, input, indices, values, reduced);
  hipLaunchKernelGGL(na_softmax_kernel, dim3(1), dim3(1024), 0, stream,
                     reduced, alpha);
}